// Abcd_41059887350013
// MI455X (gfx1250) — compile-verified
//
#include <hip/hip_runtime.h>
#include <math.h>

// Problem constants (from reference): T=730, B=1000, NMUL=8, LENF=15
// out: (T,B,6) f32. in[0]=x (T,B,2) f32, in[1]=raw_phy_static (B,34) f32.
//
// Workspace layout (floats):
//   [0, 16000)                : UH[b*16 + k]  (15 taps + zero pad), per basin
//   [16000, 16000 + 3*730000) : qsig[s*730000 + b*730 + t]  s = {Qsim,Qsurf,Qgw} means
// Total ~8.83 MB.

typedef float v2f __attribute__((ext_vector_type(2)));
typedef float v8f __attribute__((ext_vector_type(8)));

#define TLEN 730
#define BN   1000
#define RAWS 34

// ---------------------------------------------------------------------------
// Kernel 1: serial time scan, one thread per basin, NMUL=8 unrolled in regs.
// All per-cell denominators are loop-invariant -> hoisted reciprocals keep the
// 730-step dependent chain to mul/fma/sqrt/exp only.
// ---------------------------------------------------------------------------
__global__ __launch_bounds__(128) void hydro_scan_kernel(
    const float* __restrict__ x,     // (T,B,2)
    const float* __restrict__ raw,   // (B,34)
    float* __restrict__ out,         // (T,B,6) -> writes cols 3,4,5
    float* __restrict__ qsig)        // (3, B, T)
{
  int b = blockIdx.x * blockDim.x + threadIdx.x;
  if (b >= BN) return;

  float pb[8], pc[8], pd[8], S[8], G[8];
  float inv2a[8], boa[8], invb[8], invd1[8];
  const float* r = raw + b * RAWS;
#pragma unroll
  for (int m = 0; m < 8; ++m) {
    float a = r[m]      * 0.9f   + 0.1f;    // a in [0.1, 1.0]
    pb[m]   = r[8 + m]  * 450.0f + 50.0f;   // b in [50, 500]
    pc[m]   = r[16 + m];                    // c in [0, 1]
    pd[m]   = r[24 + m] * 0.89f  + 0.01f;   // d in [0.01, 0.9]
    inv2a[m] = 0.5f / a;
    boa[m]   = pb[m] / a;
    invb[m]  = 1.0f / pb[m];
    invd1[m] = 1.0f / (1.0f + pd[m]);
    S[m] = 50.0f;
    G[m] = 10.0f;
  }

  const float2* x2 = (const float2*)x;
  for (int t = 0; t < TLEN; ++t) {
    float2 f = x2[t * BN + b];               // global_load_b64 {prcp, pet}
    float p = f.x, pet = f.y;
    if (t + 4 < TLEN)
      __builtin_prefetch(&x2[(t + 4) * BN + b], 0, 0);  // global_prefetch_b8

    float qsim_s = 0.f, qsurf_s = 0.f, qgw_s = 0.f;
    float aet_s = 0.f, s_s = 0.f, g_s = 0.f;
#pragma unroll
    for (int m = 0; m < 8; ++m) {
      float W     = p + S[m];
      float term  = (W + pb[m]) * inv2a[m];
      float disc  = fmaxf(term * term - W * boa[m], 1e-5f);
      float Y     = term - sqrtf(disc);
      float Sn    = Y * expf(-pet * invb[m]);
      float AET   = Y - Sn;
      float avail = W - Y;
      float Qsurf = (1.0f - pc[m]) * avail;
      float Gn    = (G[m] + pc[m] * avail) * invd1[m];
      float Qgw   = pd[m] * Gn;
      S[m] = Sn;
      G[m] = Gn;
      qsim_s  += Qsurf + Qgw;
      qsurf_s += Qsurf;
      qgw_s   += Qgw;
      aet_s   += AET;
      s_s     += Sn;
      g_s     += Gn;
    }
    const float inv = 0.125f;  // mean over NMUL=8
    qsig[0 * (BN * TLEN) + b * TLEN + t] = qsim_s  * inv;
    qsig[1 * (BN * TLEN) + b * TLEN + t] = qsurf_s * inv;
    qsig[2 * (BN * TLEN) + b * TLEN + t] = qgw_s   * inv;
    float* o = out + (t * BN + b) * 6;
    o[3] = aet_s * inv;
    o[4] = s_s   * inv;
    o[5] = g_s   * inv;
  }
}

// ---------------------------------------------------------------------------
// Kernel 2: gamma unit hydrograph, one thread per basin. 15 taps + zero pad.
// ---------------------------------------------------------------------------
__global__ __launch_bounds__(128) void uh_gamma_kernel(
    const float* __restrict__ raw,   // (B,34)
    float* __restrict__ UH)          // (B,16)
{
  int b = blockIdx.x * blockDim.x + threadIdx.x;
  if (b >= BN) return;
  float ra = raw[b * RAWS + 32] * 2.9f;          // [0, 2.9]
  float rb = raw[b * RAWS + 33] * 6.5f;          // [0, 6.5]
  float aa = fmaxf(ra, 0.0f) + 0.1f;             // shape
  float th = fmaxf(rb, 0.0f) + 0.5f;             // scale
  float lg  = lgammaf(aa);
  float lth = logf(th);
  float w[15];
  float sum = 0.0f;
#pragma unroll
  for (int i = 0; i < 15; ++i) {
    float tm = (float)i + 0.5f;
    float lw = (aa - 1.0f) * logf(tm) - tm / th - lg - aa * lth;
    w[i] = expf(lw);
    sum += w[i];
  }
  float inv = 1.0f / sum;
#pragma unroll
  for (int i = 0; i < 15; ++i) UH[b * 16 + i] = w[i] * inv;
  UH[b * 16 + 15] = 0.0f;  // pad K to 16
}

// ---------------------------------------------------------------------------
// Kernel 3: causal per-basin convolution via V_WMMA_F32_16X16X4_F32.
// One wave = one (basin, 16-timestep tile), all 3 signals fused: the filter
// (B operand, broadcast across the 16 columns) is shared by the 3 signals, so
// each K-chunk issues 3 independent WMMAs into 3 accumulators (12 WMMAs/wave,
// no D->A/B hazards between them). A = 16x4 Toeplitz window of q (documented
// A layout: lanes 0-15 hold K pair {0,1}, lanes 16-31 hold {2,3}). D rows are
// the 16 output timesteps; lanes with N==0 (lanes 0,16) store them.
// ---------------------------------------------------------------------------
__global__ __launch_bounds__(32) void conv_wmma_kernel(
    const float* __restrict__ qsig,  // (3,B,T)
    const float* __restrict__ UH,    // (B,16)
    float* __restrict__ out)         // (T,B,6) -> writes cols 0,1,2
{
  const int NT = 46;                 // ceil(730/16), 46*16 = 736
  int wid = blockIdx.x;
  int tt  = wid % NT;
  int b   = wid / NT;
  int tbase = tt * 16;

  int L     = threadIdx.x;           // wave32 lane
  int M     = L & 15;                // output-time row handled by this lane
  int khalf = (L >> 4) * 2;          // K pair within chunk: lanes 0-15 -> {0,1}, 16-31 -> {2,3}

  const float* q0p = qsig + 0 * (BN * TLEN) + b * TLEN;
  const float* q1p = qsig + 1 * (BN * TLEN) + b * TLEN;
  const float* q2p = qsig + 2 * (BN * TLEN) + b * TLEN;
  const float* w   = UH + b * 16;

  v8f acc0 = {0.f,0.f,0.f,0.f,0.f,0.f,0.f,0.f};
  v8f acc1 = acc0, acc2 = acc0;
#pragma unroll
  for (int kk = 0; kk < 4; ++kk) {
    int k0 = kk * 4 + khalf;         // global tap index for VGPR0 of this lane
    v2f bv;
    bv.x = w[k0];
    bv.y = w[k0 + 1];
    // A[M, k] = q[tbase + M - k], zero outside [0, 729] (branchless clamp+select
    // keeps EXEC all-1s through the WMMAs).
    int t0 = tbase + M - k0;
    int t1 = t0 - 1;
    int c0 = t0 < 0 ? 0 : (t0 > TLEN - 1 ? TLEN - 1 : t0);
    int c1 = t1 < 0 ? 0 : (t1 > TLEN - 1 ? TLEN - 1 : t1);
    bool v0 = (t0 == c0), v1 = (t1 == c1);
    v2f a0, a1, a2;
    a0.x = v0 ? q0p[c0] : 0.0f;  a0.y = v1 ? q0p[c1] : 0.0f;
    a1.x = v0 ? q1p[c0] : 0.0f;  a1.y = v1 ? q1p[c1] : 0.0f;
    a2.x = v0 ? q2p[c0] : 0.0f;  a2.y = v1 ? q2p[c1] : 0.0f;
    acc0 = __builtin_amdgcn_wmma_f32_16x16x4_f32(false, a0, false, bv, (short)0, acc0, false, false);
    acc1 = __builtin_amdgcn_wmma_f32_16x16x4_f32(false, a1, false, bv, (short)0, acc1, false, false);
    acc2 = __builtin_amdgcn_wmma_f32_16x16x4_f32(false, a2, false, bv, (short)0, acc2, false, false);
  }

  // D layout: VGPR r -> (M=r, lanes 0-15) / (M=r+8, lanes 16-31); column N = lane&15.
  // All columns identical; column N==0 (lanes 0 and 16) writes the 16 rows.
  if (M == 0) {
    int half = L >> 4;
#pragma unroll
    for (int r2 = 0; r2 < 8; ++r2) {
      int t = tbase + half * 8 + r2;
      if (t < TLEN) {
        float* o = out + (t * BN + b) * 6;
        o[0] = acc0[r2];
        o[1] = acc1[r2];
        o[2] = acc2[r2];
      }
    }
  }
}

// ---------------------------------------------------------------------------
extern "C" void kernel_launch(void* const* d_in, const int* in_sizes, int n_in,
                              void* d_out, int out_size, void* d_ws, size_t ws_size,
                              hipStream_t stream) {
  (void)in_sizes; (void)n_in; (void)out_size; (void)ws_size;
  const float* x   = (const float*)d_in[0];   // (730,1000,2)
  const float* raw = (const float*)d_in[1];   // (1000,34)
  float* out = (float*)d_out;                 // (730,1000,6)

  float* wsf  = (float*)d_ws;
  float* UH   = wsf;                          // 16000 floats
  float* qsig = wsf + 16000;                  // 3*730000 floats (~8.76 MB)

  hydro_scan_kernel<<<(BN + 127) / 128, 128, 0, stream>>>(x, raw, out, qsig);
  uh_gamma_kernel  <<<(BN + 127) / 128, 128, 0, stream>>>(raw, UH);
  conv_wmma_kernel <<<BN * 46, 32, 0, stream>>>(qsig, UH, out);
}